// LogicalConsistencyLoss_72404558676612
// MI455X (gfx1250) — compile-verified
//
#include <hip/hip_runtime.h>

typedef __attribute__((ext_vector_type(2))) float v2f;
typedef __attribute__((ext_vector_type(8))) float v8f;

#define Nn 512
#define Rr 4
#define Bb 2
#define TA 32            // a-tile per workgroup
#define TB 64            // b-tile per workgroup
#define CC 128           // c chunk staged in LDS
#define PA_STRIDE 33     // 32 + 1 pad (bank-conflict-free)
#define PB_STRIDE 65     // 64 + 1 pad
#define PA_OFF 0
#define PA_SIZE (CC * PA_STRIDE)            // 4224 floats
#define PB_OFF PA_SIZE
#define PB_SIZE (CC * PB_STRIDE)            // 8320 floats
#define ZERO_IDX (PA_OFF + PA_SIZE + PB_SIZE)
#define RED_OFF (ZERO_IDX + 1)
#define LDS_FLOATS (RED_OFF + 256)          // ~51.2 KB

__device__ __forceinline__ float sigmoidf_(float x) {
  return 1.0f / (1.0f + __expf(-x));
}

__global__ __launch_bounds__(256)
void trans_loss_kernel(const float* __restrict__ logits,
                       const int* __restrict__ masks,
                       float* __restrict__ partials) {
  __shared__ float lds[LDS_FLOATS];
  const int tid  = threadIdx.x;
  const int aBlk = blockIdx.x;   // 0..15
  const int bBlk = blockIdx.y;   // 0..7
  const int br   = blockIdx.z;   // 0..7
  const int bb = br >> 2;
  const int r  = br & 3;
  const int a0 = aBlk * TA;
  const int b0 = bBlk * TB;

  const int lane = tid & 31;
  const int w    = tid >> 5;
  const int aSub = w >> 2;       // 0..1
  const int bSub = w & 3;        // 0..3
  const bool low = lane < 16;
  const int mbase = bb * Nn;

  // ---- load this wave's 16x16 rel tile into registers, WMMA C/D layout ----
  // lane l, vgpr v  ->  M = v + 8*(l>=16), N = l&15
  const int aT = a0 + aSub * 16 + ((lane >> 4) << 3);
  const int bT = b0 + bSub * 16 + (lane & 15);
  const float mB = (masks[mbase + bT] > 0) ? 1.0f : 0.0f;
  float rel[8];
#pragma unroll
  for (int v = 0; v < 8; ++v) {
    const int a = aT + v;
    const float mA = (masks[mbase + a] > 0) ? 1.0f : 0.0f;
    const float x = logits[(((size_t)bb * Nn + a) * Nn + bT) * Rr + r];
    rel[v] = sigmoidf_(x) * mA * mB;
  }

  if (tid == 0) lds[ZERO_IDX] = 0.0f;   // zero slot read by lanes 16..31

  float acc[8];
#pragma unroll
  for (int v = 0; v < 8; ++v) acc[v] = 0.0f;

  // per-lane LDS walk: low lanes stride through the column arrays,
  // high lanes pin to the zero slot (stride 0) -> WMMA K=1..3 operands are 0.
  const int ia0 = low ? (PA_OFF + aSub * 16 + lane) : ZERO_IDX;
  const int da  = low ? PA_STRIDE : 0;
  const int ib0 = low ? (PB_OFF + bSub * 16 + lane) : ZERO_IDX;
  const int db  = low ? PB_STRIDE : 0;

  const v8f zeroC = {0.f, 0.f, 0.f, 0.f, 0.f, 0.f, 0.f, 0.f};

  for (int cc = 0; cc < Nn; cc += CC) {
    __syncthreads();  // previous chunk fully consumed

    // stage p~[a_tile, c_chunk]: 32 rows x 128 c  (masked sigmoid)
#pragma unroll
    for (int k = 0; k < (TA * CC) / 256; ++k) {
      const int idx = k * 256 + tid;
      const int i = idx >> 7;           // row within tile
      const int c = idx & (CC - 1);
      const int row = a0 + i;
      const float mR = (masks[mbase + row] > 0) ? 1.0f : 0.0f;
      const float mC = (masks[mbase + cc + c] > 0) ? 1.0f : 0.0f;
      const float x = logits[(((size_t)bb * Nn + row) * Nn + (cc + c)) * Rr + r];
      lds[PA_OFF + c * PA_STRIDE + i] = sigmoidf_(x) * mR * mC;
    }
    // stage p~[b_tile, c_chunk]: 64 rows x 128 c
#pragma unroll
    for (int k = 0; k < (TB * CC) / 256; ++k) {
      const int idx = k * 256 + tid;
      const int j = idx >> 7;
      const int c = idx & (CC - 1);
      const int row = b0 + j;
      const float mR = (masks[mbase + row] > 0) ? 1.0f : 0.0f;
      const float mC = (masks[mbase + cc + c] > 0) ? 1.0f : 0.0f;
      const float x = logits[(((size_t)bb * Nn + row) * Nn + (cc + c)) * Rr + r];
      lds[PB_OFF + c * PB_STRIDE + j] = sigmoidf_(x) * mR * mC;
    }
    __syncthreads();

    int ia = ia0, ib = ib0;
#pragma unroll 4
    for (int c = 0; c < CC; ++c) {
      const float av = lds[ia];
      const float bv = lds[ib];
      ia += da; ib += db;
      // q[a,b] = p~[a,c] * p~[b,c] via 16x16x4 f32 WMMA with only K=0 live
      v2f A  = {av, 0.0f};
      v2f Bm = {bv, 0.0f};
      v8f q = __builtin_amdgcn_wmma_f32_16x16x4_f32(
          /*neg_a=*/false, A, /*neg_b=*/false, Bm,
          /*c_mod=*/(short)0, zeroC, /*reuse_a=*/false, /*reuse_b=*/false);
#pragma unroll
      for (int v = 0; v < 8; ++v) {
        float t = rel[v] - q[v];
        acc[v] += fmaxf(t, 0.0f);
      }
    }
  }

  // ---- deterministic in-workgroup reduction ----
  float s = 0.0f;
#pragma unroll
  for (int v = 0; v < 8; ++v) s += acc[v];
  __syncthreads();
  lds[RED_OFF + tid] = s;
  __syncthreads();
  for (int off = 128; off > 0; off >>= 1) {
    if (tid < off) lds[RED_OFF + tid] += lds[RED_OFF + tid + off];
    __syncthreads();
  }
  if (tid == 0) {
    const int wg = (blockIdx.z * gridDim.y + blockIdx.y) * gridDim.x + blockIdx.x;
    partials[wg] = lds[RED_OFF];
  }
}

__global__ __launch_bounds__(256)
void final_reduce_kernel(const float* __restrict__ partials,
                         float* __restrict__ out, int n) {
  __shared__ float red[256];
  const int tid = threadIdx.x;
  float s = 0.0f;
  for (int i = tid; i < n; i += 256) s += partials[i];
  red[tid] = s;
  __syncthreads();
  for (int off = 128; off > 0; off >>= 1) {
    if (tid < off) red[tid] += red[tid + off];
    __syncthreads();
  }
  if (tid == 0) out[0] = red[0] * (1.0f / (float)(Rr * Bb));  // WEIGHT*sum/(R*B)
}

extern "C" void kernel_launch(void* const* d_in, const int* in_sizes, int n_in,
                              void* d_out, int out_size, void* d_ws, size_t ws_size,
                              hipStream_t stream) {
  (void)in_sizes; (void)n_in; (void)out_size; (void)ws_size;
  const float* logits = (const float*)d_in[0];   // [B, N, N, R] f32
  const int*   masks  = (const int*)d_in[1];     // [B, N] i32
  float* out = (float*)d_out;                    // scalar f32
  float* partials = (float*)d_ws;                // 1024 floats (4 KB)

  dim3 grid(Nn / TA, Nn / TB, Bb * Rr);          // 16 x 8 x 8 = 1024 WGs
  trans_loss_kernel<<<grid, 256, 0, stream>>>(logits, masks, partials);
  final_reduce_kernel<<<1, 256, 0, stream>>>(partials, out,
                                             (Nn / TA) * (Nn / TB) * Bb * Rr);
}